// ManifoldEmbedding_75041668596321
// MI455X (gfx1250) — compile-verified
//
#include <hip/hip_runtime.h>
#include <hip/hip_bf16.h>

typedef __attribute__((ext_vector_type(2))) float v2f;
typedef __attribute__((ext_vector_type(8))) float v8f;

#define ME_EPS 1e-7f
#define N_PTS 4096
#define D_E   32
#define D_H   33

// ---------------------------------------------------------------------------
// Prep: row squared-norms of x_euc, plus softplus of the three scale scalars.
// ---------------------------------------------------------------------------
__global__ __launch_bounds__(256) void me_prep(const float* __restrict__ xe,
                                               const float* __restrict__ s0,
                                               const float* __restrict__ s1,
                                               const float* __restrict__ s2,
                                               float* __restrict__ sqe,
                                               float* __restrict__ scal) {
  int i = blockIdx.x * blockDim.x + threadIdx.x;
  if (i < N_PTS) {
    const float* row = xe + (size_t)i * D_E;
    float acc = 0.f;
#pragma unroll
    for (int d = 0; d < D_E; ++d) acc = fmaf(row[d], row[d], acc);
    sqe[i] = acc;
  }
  if (i < 3) {
    float s = (i == 0) ? s0[0] : ((i == 1) ? s1[0] : s2[0]);
    // numerically stable softplus: max(s,0) + log1p(exp(-|s|))
    scal[i] = fmaxf(s, 0.f) + log1pf(expf(-fabsf(s)));
  }
}

// ---------------------------------------------------------------------------
// Main: one 16x16 output tile per wave32. Three fp32 WMMA Gram accumulations
// (euclidean / sphere / lorentz-spatial), then transcendental epilogue.
//
// A-fragment layout for V_WMMA_F32_16X16X4_F32 (ISA 7.12.2, 32-bit A 16x4):
//   lane L in [0,16): row M=L,    holds K=kb+0 (v0), kb+1 (v1)
//   lane L in [16,32): row M=L-16, holds K=kb+2 (v0), kb+3 (v1)
// => per-lane float2 load at X[(i0 + (L&15))*stride + kb + 2*(L>>4)].
// B (4x16) mirrors this with the column-tile rows, since G = X * X^T.
// C/D layout: VGPR r, lane L -> (row = r + 8*(L>>4), col = L&15).
// ---------------------------------------------------------------------------
__global__ __launch_bounds__(256) void me_tile(const float* __restrict__ xe,
                                               const float* __restrict__ xs,
                                               const float* __restrict__ xh,
                                               const float* __restrict__ sqe,
                                               const float* __restrict__ scal,
                                               float* __restrict__ out) {
  const int lane = threadIdx.x & 31;
  const int wave = threadIdx.x >> 5;
  const int tile = blockIdx.x * 8 + wave;   // 65536 tiles total
  const int i0 = (tile >> 8) << 4;          // output row base
  const int j0 = (tile & 255) << 4;         // output col base
  const int lo = lane & 15;
  const int hi = lane >> 4;

  const float* ae = xe + (size_t)(i0 + lo) * D_E + 2 * hi;
  const float* be = xe + (size_t)(j0 + lo) * D_E + 2 * hi;
  const float* as = xs + (size_t)(i0 + lo) * D_E + 2 * hi;
  const float* bs = xs + (size_t)(j0 + lo) * D_E + 2 * hi;
  const float* ah = xh + (size_t)(i0 + lo) * D_H + 1 + 2 * hi;  // spatial part
  const float* bh = xh + (size_t)(j0 + lo) * D_H + 1 + 2 * hi;

  v8f ge = {}, gs = {}, gh = {};
#pragma unroll
  for (int kb = 0; kb < D_E; kb += 4) {
    v2f a0 = *(const v2f*)(ae + kb);
    v2f b0 = *(const v2f*)(be + kb);
    ge = __builtin_amdgcn_wmma_f32_16x16x4_f32(false, a0, false, b0,
                                               (short)0, ge, false, false);
    v2f a1 = *(const v2f*)(as + kb);
    v2f b1 = *(const v2f*)(bs + kb);
    gs = __builtin_amdgcn_wmma_f32_16x16x4_f32(false, a1, false, b1,
                                               (short)0, gs, false, false);
    // Lorentz spatial rows have odd stride (33) + offset 1 -> scalar loads
    v2f a2; a2.x = ah[kb]; a2.y = ah[kb + 1];
    v2f b2; b2.x = bh[kb]; b2.y = bh[kb + 1];
    gh = __builtin_amdgcn_wmma_f32_16x16x4_f32(false, a2, false, b2,
                                               (short)0, gh, false, false);
  }

  const float sp0 = scal[0], sp1 = scal[1], sp2 = scal[2];
  const float sq_j = sqe[j0 + lo];
  const float x0_j = xh[(size_t)(j0 + lo) * D_H];
  const int mbase = i0 + 8 * hi;
  float* orow = out + (size_t)mbase * N_PTS + j0 + lo;

#pragma unroll
  for (int r = 0; r < 8; ++r) {
    const int m = mbase + r;
    const float sq_i = sqe[m];          // broadcast within half-wave (cache hit)
    const float x0_i = xh[(size_t)m * D_H];

    // Euclidean: ||i||^2 + ||j||^2 - 2<i,j>, clamped at 0
    float d2e = fmaxf(fmaf(-2.f, ge[r], sq_i + sq_j), 0.f);

    // Sphere: acos(clip(<i,j>))^2
    float c = fminf(fmaxf(gs[r], -1.f + ME_EPS), 1.f - ME_EPS);
    float dsv = acosf(c);

    // Lorentz: z = max(x0_i*x0_j - <sp_i,sp_j>, 1+eps); acosh(z)^2
    float z = fmaxf(fmaf(x0_i, x0_j, -gh[r]), 1.f + ME_EPS);
    float dhv = acoshf(z);

    orow[(size_t)r * N_PTS] =
        fmaf(sp0, d2e, fmaf(sp1, dsv * dsv, sp2 * (dhv * dhv)));
  }
}

// ---------------------------------------------------------------------------
extern "C" void kernel_launch(void* const* d_in, const int* in_sizes, int n_in,
                              void* d_out, int out_size, void* d_ws, size_t ws_size,
                              hipStream_t stream) {
  const float* xe = (const float*)d_in[0];  // x_euc  4096x32
  const float* xs = (const float*)d_in[1];  // x_sph  4096x32
  const float* xh = (const float*)d_in[2];  // x_hyp  4096x33
  const float* s0 = (const float*)d_in[3];
  const float* s1 = (const float*)d_in[4];
  const float* s2 = (const float*)d_in[5];
  float* out  = (float*)d_out;              // 4096x4096 f32
  float* sqe  = (float*)d_ws;               // 4096 norms
  float* scal = sqe + N_PTS;                // 3 softplus scalars

  me_prep<<<(N_PTS + 255) / 256, 256, 0, stream>>>(xe, s0, s1, s2, sqe, scal);

  const int tiles = (N_PTS / 16) * (N_PTS / 16);     // 65536
  me_tile<<<tiles / 8, 256, 0, stream>>>(xe, xs, xh, sqe, scal, out);
}